// VGGBlock_9113920602419
// MI455X (gfx1250) — compile-verified
//
#include <hip/hip_runtime.h>
#include <hip/hip_bf16.h>
#include <math.h>

typedef __attribute__((ext_vector_type(2))) float v2f;
typedef __attribute__((ext_vector_type(8))) float v8f;

#define NCH 128
#define BN_EPS 1e-5f
#define ELU_ALPHA 0.1f
#define LOOP_W 2.0f

// ---------- monotone float<->uint order-preserving encoding (for atomic max pool) ----------
__device__ __forceinline__ unsigned ordenc(float f) {
  unsigned u = __float_as_uint(f);
  return (u & 0x80000000u) ? ~u : (u | 0x80000000u);
}
__device__ __forceinline__ float orddec(unsigned k) {
  unsigned u = (k & 0x80000000u) ? (k & 0x7fffffffu) : ~k;
  return __uint_as_float(u);
}

// LDS swizzle for transposed W: keeps (k,k+1) pairs contiguous & even-aligned,
// spreads half-wave accesses across bank pairs.
__device__ __forceinline__ int wswz(int n, int k) {
  return n * NCH + (k ^ ((n & 7) << 3));
}

// ---------- degree / norm ----------
__global__ __launch_bounds__(256) void k_deg_init(float* deg, int n) {
  int i = blockIdx.x * 256 + threadIdx.x;
  if (i < n) deg[i] = LOOP_W;  // self-loop weight
}

__global__ __launch_bounds__(256) void k_deg_edges(float* deg, const int* dst,
                                                   const float* w, int nE) {
  int e = blockIdx.x * 256 + threadIdx.x;
  if (e < nE) atomicAdd(&deg[dst[e]], w[e]);
}

__global__ __launch_bounds__(256) void k_dinv(const float* deg, float* dinv, int n) {
  int i = blockIdx.x * 256 + threadIdx.x;
  if (i < n) {
    float d = deg[i];
    dinv[i] = (d > 0.f) ? rsqrtf(d) : 0.f;
  }
}

// ---------- fp32 WMMA GEMM: C[M x 128] = f(A)[M x 128] @ W[128 x 128] ----------
// f(A) = A*scale[ch] + shift[ch] when HAS_BN (fused BN of the previous layer).
// One wave per 16-row strip; 8 accumulator tiles across N; K loop in steps of 4.
// W is staged transposed+swizzled in LDS so each B fragment is one ds_load_b64.
template <bool HAS_BN>
__global__ __launch_bounds__(256) void k_gemm_wmma(
    const float* __restrict__ A, const float* __restrict__ W,
    float* __restrict__ C, const float* __restrict__ scale,
    const float* __restrict__ shift, int M) {
  __shared__ float Wt[NCH * NCH];  // 64 KB: Wt[wswz(n,k)] = W[k][n]
  for (int i = threadIdx.x; i < NCH * NCH; i += 256) {
    int k = i >> 7, n = i & (NCH - 1);  // coalesced global read of W[k][n]
    Wt[wswz(n, k)] = W[i];
  }
  __syncthreads();

  const int wave = threadIdx.x >> 5;
  const int lane = threadIdx.x & 31;
  const int strip = blockIdx.x * 8 + wave;
  if (strip * 16 >= M) return;
  const int r0 = strip * 16;
  const int mrow = lane & 15;          // A-frag M index
  const int khalf = (lane >> 4) << 1;  // 0 or 2: K sub-offset per lane half
  const int ncol = lane & 15;          // B-frag N index within tile

  v8f acc[8] = {};
  const float* Arow = A + (size_t)(r0 + mrow) * NCH;

#pragma unroll 4
  for (int kk = 0; kk < NCH; kk += 4) {
    const int kb = kk + khalf;
    v2f a = *(const v2f*)&Arow[kb];  // contiguous 8B global load
    if (HAS_BN) {
      a.x = a.x * scale[kb] + shift[kb];
      a.y = a.y * scale[kb + 1] + shift[kb + 1];
    }
#pragma unroll
    for (int t = 0; t < 8; ++t) {
      const int n = t * 16 + ncol;
      const v2f b = *(const v2f*)&Wt[wswz(n, kb)];  // one ds_load_b64
      acc[t] = __builtin_amdgcn_wmma_f32_16x16x4_f32(
          false, a, false, b, (short)0, acc[t], false, false);
    }
  }

  const int colbase = lane & 15;
  const int rowoff = (lane >> 4) * 8;
#pragma unroll
  for (int t = 0; t < 8; ++t) {
#pragma unroll
    for (int r = 0; r < 8; ++r) {
      int row = r0 + rowoff + r;
      if (row < M) C[(size_t)row * NCH + t * 16 + colbase] = acc[t][r];
    }
  }
}

// ---------- aggregation ----------
// out[i][c] = bias[c] + h[i][c] * (LOOP_W * dinv[i]^2)   (self-loop + bias init)
__global__ __launch_bounds__(256) void k_agg_init(
    float* __restrict__ out, const float* __restrict__ h,
    const float* __restrict__ dinv, const float* __restrict__ bias, int n) {
  size_t i = (size_t)blockIdx.x * 256 + threadIdx.x;
  if (i >= (size_t)n * NCH) return;
  int ch = (int)(i & (NCH - 1));
  int node = (int)(i >> 7);
  float dv = dinv[node];
  out[i] = bias[ch] + h[i] * (LOOP_W * dv * dv);
}

// one wave32 per edge, float4 per lane: coalesced 512B row gather + 4 L2 atomics/lane
__global__ __launch_bounds__(256) void k_agg_edges(
    float* __restrict__ out, const float* __restrict__ h,
    const float* __restrict__ dinv, const int* __restrict__ src,
    const int* __restrict__ dst, const float* __restrict__ w, int nE) {
  int e = blockIdx.x * 8 + (threadIdx.x >> 5);
  if (e >= nE) return;
  int lane = threadIdx.x & 31;
  int s = src[e], d = dst[e];
  float norm = dinv[s] * w[e] * dinv[d];
  const float4 hv = *(const float4*)&h[(size_t)s * NCH + lane * 4];
  float* o = &out[(size_t)d * NCH + lane * 4];
  atomicAdd(o + 0, hv.x * norm);
  atomicAdd(o + 1, hv.y * norm);
  atomicAdd(o + 2, hv.z * norm);
  atomicAdd(o + 3, hv.w * norm);
}

// ---------- batch norm (training stats) ----------
__global__ void k_zero256(float* p) { p[threadIdx.x] = 0.f; }

__global__ __launch_bounds__(128) void k_bn_stats(
    const float* __restrict__ h, float* __restrict__ sums,
    float* __restrict__ sumsq, int n) {
  int c = threadIdx.x;  // channel
  int r0 = blockIdx.x * 256;
  int r1 = r0 + 256;
  if (r1 > n) r1 = n;
  float s = 0.f, q = 0.f;
  for (int r = r0; r < r1; ++r) {
    float v = h[(size_t)r * NCH + c];
    s += v;
    q += v * v;
  }
  atomicAdd(&sums[c], s);
  atomicAdd(&sumsq[c], q);
}

__global__ void k_bn_finalize(const float* sums, const float* sumsq,
                              const float* gamma, const float* beta,
                              float* a, float* c, float invN) {
  int ch = threadIdx.x;  // 128 threads
  float mu = sums[ch] * invN;
  float var = sumsq[ch] * invN - mu * mu;
  float inv = rsqrtf(var + BN_EPS);
  float ga = gamma[ch] * inv;
  a[ch] = ga;
  c[ch] = beta[ch] - mu * ga;
}

// ---------- BN2-apply + encode keys for max pool ----------
// y[i] = agg[i]*a[ch]+c[ch]; write y to ybuf; write ordenc(y) over key (aliases agg)
__global__ __launch_bounds__(256) void k_bn_apply_enc(
    const float* agg, float* __restrict__ ybuf, unsigned* key,
    const float* __restrict__ a, const float* __restrict__ c, int n) {
  size_t i = (size_t)blockIdx.x * 256 + threadIdx.x;
  if (i >= (size_t)n * NCH) return;
  int ch = (int)(i & (NCH - 1));
  float v = agg[i] * a[ch] + c[ch];
  ybuf[i] = v;
  key[i] = ordenc(v);  // self term of the max pool
}

__global__ __launch_bounds__(256) void k_pool_edges(
    unsigned* __restrict__ key, const float* __restrict__ y,
    const int* __restrict__ src, const int* __restrict__ dst, int nE) {
  int e = blockIdx.x * 8 + (threadIdx.x >> 5);
  if (e >= nE) return;
  int lane = threadIdx.x & 31;
  int s = src[e], d = dst[e];
  const float4 yv = *(const float4*)&y[(size_t)s * NCH + lane * 4];
  unsigned* o = &key[(size_t)d * NCH + lane * 4];
  atomicMax(o + 0, ordenc(yv.x));
  atomicMax(o + 1, ordenc(yv.y));
  atomicMax(o + 2, ordenc(yv.z));
  atomicMax(o + 3, ordenc(yv.w));
}

__global__ __launch_bounds__(256) void k_decode_elu(unsigned* keyin, float* out, int n) {
  size_t i = (size_t)blockIdx.x * 256 + threadIdx.x;
  if (i >= (size_t)n * NCH) return;
  float v = orddec(keyin[i]);
  out[i] = (v > 0.f) ? v : ELU_ALPHA * (expf(v) - 1.f);
}

// ---------- host ----------
extern "C" void kernel_launch(void* const* d_in, const int* in_sizes, int n_in,
                              void* d_out, int out_size, void* d_ws, size_t ws_size,
                              hipStream_t stream) {
  const float* x  = (const float*)d_in[0];
  const int* ei   = (const int*)d_in[1];
  const float* ew = (const float*)d_in[2];
  const float* W1 = (const float*)d_in[3];
  const float* b1 = (const float*)d_in[4];
  const float* g1 = (const float*)d_in[5];
  const float* be1 = (const float*)d_in[6];
  const float* W2 = (const float*)d_in[7];
  const float* b2 = (const float*)d_in[8];
  const float* g2 = (const float*)d_in[9];
  const float* be2 = (const float*)d_in[10];

  const int n = in_sizes[0] / NCH;   // 100000 nodes
  const int E = in_sizes[2];         // 1600000 edges
  const int* src = ei;
  const int* dst = ei + E;

  char* ws = (char*)d_ws;
  float* hbuf  = (float*)ws;                               // n*128 f32 = 51.2MB
  float* deg   = (float*)(ws + (size_t)n * NCH * 4);
  float* dinv  = deg + n;
  float* sums  = dinv + n;
  float* sumsq = sums + 128;
  float* abn   = sumsq + 128;
  float* cbn   = abn + 128;

  float* outf = (float*)d_out;        // n*128 f32, ping-pong vs hbuf
  unsigned* outu = (unsigned*)d_out;

  const int gN   = (n + 255) / 256;                // node-wise
  const int gNC  = (n * NCH + 255) / 256;          // element-wise
  const int gE1  = (E + 255) / 256;                // edge per thread
  const int gE8  = (E + 7) / 8;                    // edge per wave
  const int gBN  = (n + 255) / 256;                // 256 rows per block
  const int gGEMM = (n / 16 + 7) / 8;              // 8 strips (waves) per block
  const float invN = 1.0f / (float)n;

  // degrees + normalization
  k_deg_init<<<gN, 256, 0, stream>>>(deg, n);
  k_deg_edges<<<gE1, 256, 0, stream>>>(deg, dst, ew, E);
  k_dinv<<<gN, 256, 0, stream>>>(deg, dinv, n);

  // ---- layer 1: h1 = x @ W1 -> hbuf ; agg1 -> d_out
  k_gemm_wmma<false><<<gGEMM, 256, 0, stream>>>(x, W1, hbuf, nullptr, nullptr, n);
  k_agg_init<<<gNC, 256, 0, stream>>>(outf, hbuf, dinv, b1, n);
  k_agg_edges<<<gE8, 256, 0, stream>>>(outf, hbuf, dinv, src, dst, ew, E);
  k_zero256<<<1, 256, 0, stream>>>(sums);  // zeros sums[128] + sumsq[128] (contiguous)
  k_bn_stats<<<gBN, 128, 0, stream>>>(outf, sums, sumsq, n);
  k_bn_finalize<<<1, 128, 0, stream>>>(sums, sumsq, g1, be1, abn, cbn, invN);

  // ---- layer 2: h2 = bn1(agg1) @ W2 -> hbuf ; agg2 -> d_out
  k_gemm_wmma<true><<<gGEMM, 256, 0, stream>>>(outf, W2, hbuf, abn, cbn, n);
  k_agg_init<<<gNC, 256, 0, stream>>>(outf, hbuf, dinv, b2, n);
  k_agg_edges<<<gE8, 256, 0, stream>>>(outf, hbuf, dinv, src, dst, ew, E);
  k_zero256<<<1, 256, 0, stream>>>(sums);
  k_bn_stats<<<gBN, 128, 0, stream>>>(outf, sums, sumsq, n);
  k_bn_finalize<<<1, 128, 0, stream>>>(sums, sumsq, g2, be2, abn, cbn, invN);

  // ---- bn2 apply -> y(hbuf) + encoded self keys over d_out; neighbor max; decode+ELU
  k_bn_apply_enc<<<gNC, 256, 0, stream>>>(outf, hbuf, outu, abn, cbn, n);
  k_pool_edges<<<gE8, 256, 0, stream>>>(outu, hbuf, src, dst, E);
  k_decode_elu<<<gNC, 256, 0, stream>>>(outu, outf, n);
}